// SpatioTemporalEGNN_5119601017285
// MI455X (gfx1250) — compile-verified
//
#include <hip/hip_runtime.h>

// SpatioTemporalEGNN forward, fused for MI455X (gfx1250, wave32, WMMA).
// B=64, N=100, NF=16, EF=8, HID=64, L=4. Edges are sender-major contiguous.

#define B_    64
#define N_    100
#define NF_   16
#define EF_   8
#define HID_  64
#define L_    4
#define DEG_  (N_ - 1)          // 99 edges per sender node
#define E_    (N_ * DEG_)       // 9900 edges per batch graph
#define NN_   (B_ * N_)         // 6400 nodes total
#define MPAD_ 112               // 99 edges padded to 7 tiles of 16
#define LDA_  66                // LDS row stride (words) to dodge bank conflicts

typedef float v2f __attribute__((ext_vector_type(2)));
typedef float v8f __attribute__((ext_vector_type(8)));
typedef int   v4i __attribute__((ext_vector_type(4)));

#if defined(__has_builtin)
#if __has_builtin(__builtin_amdgcn_global_load_async_to_lds_b128) && \
    __has_builtin(__builtin_amdgcn_s_wait_asynccnt)
#define USE_ASYNC_LDS 1
#endif
#endif
#ifndef USE_ASYNC_LDS
#define USE_ASYNC_LDS 0
#endif

__device__ __forceinline__ float silu_f(float v) {
  return v * (1.0f / (1.0f + __expf(-v)));
}

#if USE_ASYNC_LDS
// One 16-byte chunk, global -> LDS, via the CDNA5 async-to-LDS DMA path.
// Builtin parameter type (from hipcc diagnostic): int __vector(4) pointers.
__device__ __forceinline__ void async_cp16(const float* g, float* l) {
  __builtin_amdgcn_global_load_async_to_lds_b128(
      (v4i*)g,
      (__attribute__((address_space(3))) v4i*)l, 0, 0);
}
#endif

// ---------------------------------------------------------------------------
// hh0 = h @ emb_W + emb_b ; x0 = rot_rep (copy). One block per node, 64 thr.
// ---------------------------------------------------------------------------
__global__ void __launch_bounds__(64)
egnn_init(const float* __restrict__ hfeat, const float* __restrict__ rot,
          const float* __restrict__ embW, const float* __restrict__ embB,
          float* __restrict__ hh, float* __restrict__ x) {
  const int i = blockIdx.x, h = threadIdx.x;
  float acc = embB[h];
#pragma unroll
  for (int f = 0; f < NF_; ++f) acc += hfeat[i * NF_ + f] * embW[f * HID_ + h];
  hh[i * HID_ + h] = acc;
  if (h < 6) x[i * 6 + h] = rot[i * 6 + h];
}

// ---------------------------------------------------------------------------
// Per-layer projection: QP[n][0:64]   = hh[n] @ W1[0:64]    (hh[row] part)
//                       QP[n][64:128] = hh[n] @ W1[64:128]  (hh[col] part)
// M = 6400, K = 64, N = 128. 400 blocks x 8 waves; one 16x16 tile per wave.
// ---------------------------------------------------------------------------
__global__ void __launch_bounds__(256)
egnn_proj(const float* __restrict__ hh, const float* __restrict__ W1,
          float* __restrict__ QP) {
  const int tid  = threadIdx.x;
  const int wave = tid >> 5;          // 0..7 -> N tile
  const int lane = tid & 31;
  const int half = lane >> 4;         // 0: K=k..k+1 / M=r ; 1: K=k+2..k+3 / M=8+r
  const int l16  = lane & 15;
  const int rowBase = blockIdx.x * 16;
  const int j = wave * 16 + l16;      // output column 0..127
  const int rowOff = (j >= HID_) ? HID_ : 0;
  const int jj = j & (HID_ - 1);

  v8f acc = {0.f, 0.f, 0.f, 0.f, 0.f, 0.f, 0.f, 0.f};
  const float* aRow = hh + (size_t)(rowBase + l16) * HID_;
#pragma unroll
  for (int k = 0; k < HID_; k += 4) {
    const int ka = k + half * 2;
    v2f a, b;
    a.x = aRow[ka];
    a.y = aRow[ka + 1];
    b.x = W1[(rowOff + ka) * HID_ + jj];
    b.y = W1[(rowOff + ka + 1) * HID_ + jj];
    acc = __builtin_amdgcn_wmma_f32_16x16x4_f32(false, a, false, b,
                                                (short)0, acc, false, false);
  }
#pragma unroll
  for (int r = 0; r < 8; ++r) {
    const int m = rowBase + r + 8 * half;
    QP[(size_t)m * 128 + j] = acc[r];
  }
}

// ---------------------------------------------------------------------------
// Fused per-node layer: edge MLP + coord MLP + aggregation + node MLP.
// One block (256 thr = 8 waves) per sender node; its 99 edges are contiguous.
// Both weight matrices are staged once (async DMA to LDS when available) and
// each wave keeps a fixed output-column tile so B fragments live in VGPRs.
// ---------------------------------------------------------------------------
__global__ void __launch_bounds__(256)
egnn_layer(const float* __restrict__ hh_in, const float* __restrict__ x_in,
           const float* __restrict__ QP,    const float* __restrict__ eattr,
           const float* __restrict__ W1,    const float* __restrict__ b1,
           const float* __restrict__ W2,    const float* __restrict__ b2,
           const float* __restrict__ nW1,   const float* __restrict__ nb1,
           const float* __restrict__ nW2,   const float* __restrict__ nb2,
           const float* __restrict__ C1,    const float* __restrict__ cb1,
           const float* __restrict__ C2,    const float* __restrict__ cb2,
           float* __restrict__ hh_out, float* __restrict__ x_out) {
  __shared__ __align__(16) float s_a1[MPAD_ * LDA_];  // a1, reused for coord hidden
  __shared__ __align__(16) float s_m [MPAD_ * LDA_];  // edge messages m
  __shared__ __align__(16) float s_w2[HID_ * HID_];   // edge W2
  __shared__ __align__(16) float s_c1[HID_ * HID_];   // coord W1
  __shared__ __align__(16) float s_dr[MPAD_ * 8];     // per-edge diff(6)+radial(2)
  __shared__ __align__(16) float s_ea[DEG_ * EF_];    // per-edge attributes
  __shared__ __align__(16) float s_w1t[11 * HID_];    // W1 rows 128..137 + b1
  __shared__ float s_q[HID_], s_hh[HID_], s_agg[HID_], s_u[HID_];
  __shared__ float s_xi[6], s_xacc[6];

  const int i = blockIdx.x;
  const int b = i / N_, n = i % N_;
  const int tid  = threadIdx.x;
  const int wave = tid >> 5;
  const int lane = tid & 31;
  const int half = lane >> 4;
  const int l16  = lane & 15;

  const float* eaN = eattr + ((size_t)b * E_ + (size_t)n * DEG_) * EF_;

  // ---- phase 0: stage weights / per-edge scalars ----
#if USE_ASYNC_LDS
  // Kick off all LDS staging through the async DMA path first so it overlaps
  // the VALU work below. ASYNCcnt is drained once before the first barrier.
  for (int c = tid; c < (HID_ * HID_) / 4; c += 256)        // W2: 1024 chunks
    async_cp16(W2 + 4 * c, s_w2 + 4 * c);
  for (int c = tid; c < (HID_ * HID_) / 4; c += 256)        // C1: 1024 chunks
    async_cp16(C1 + 4 * c, s_c1 + 4 * c);
  if (tid < (DEG_ * EF_) / 4)                               // eattr: 198 chunks
    async_cp16(eaN + 4 * tid, s_ea + 4 * tid);
  if (tid < (10 * HID_) / 4)                                // W1 tail: 160 chunks
    async_cp16(W1 + 128 * HID_ + 4 * tid, s_w1t + 4 * tid);
  if (tid < HID_ / 4)                                       // b1: 16 chunks
    async_cp16(b1 + 4 * tid, s_w1t + 10 * HID_ + 4 * tid);
#else
  for (int idx = tid; idx < HID_ * HID_; idx += 256) s_w2[idx] = W2[idx];
  for (int idx = tid; idx < HID_ * HID_; idx += 256) s_c1[idx] = C1[idx];
  for (int idx = tid; idx < DEG_ * EF_; idx += 256) s_ea[idx] = eaN[idx];
  for (int idx = tid; idx < 11 * HID_; idx += 256) {
    const int r = idx >> 6, hc = idx & 63;
    s_w1t[idx] = (r < 10) ? W1[(128 + r) * HID_ + hc] : b1[hc];
  }
#endif
  if (tid < HID_) {
    s_q [tid] = QP[(size_t)i * 128 + tid];       // hh[row]@W1_top, const per node
    s_hh[tid] = hh_in[(size_t)i * HID_ + tid];
  }
  if (tid < 6) { s_xi[tid] = x_in[(size_t)i * 6 + tid]; s_xacc[tid] = 0.f; }
  if (tid < DEG_) {                      // diff = x[row] - x[col], radial
    const int e = tid;
    const int j = (e < n) ? e : e + 1;
    const float* xr = x_in + (size_t)i * 6;
    const float* xc = x_in + (size_t)(b * N_ + j) * 6;
    float r0 = 0.f, r1 = 0.f;
#pragma unroll
    for (int d = 0; d < 3; ++d) {
      const float d0 = xr[d * 2 + 0] - xc[d * 2 + 0];
      const float d1 = xr[d * 2 + 1] - xc[d * 2 + 1];
      s_dr[e * 8 + d * 2 + 0] = d0;
      s_dr[e * 8 + d * 2 + 1] = d1;
      r0 += d0 * d0; r1 += d1 * d1;
    }
    s_dr[e * 8 + 6] = r0; s_dr[e * 8 + 7] = r1;
  }
#if USE_ASYNC_LDS
  __builtin_amdgcn_s_wait_asynccnt(0);
#endif
  __syncthreads();

  // ---- phase 1: a1 = silu(Q[i] + P[col] + radial/ea terms + b1) ----
  {
    const int h = tid & 63;
    for (int e = tid >> 6; e < MPAD_; e += 4) {
      float val = 0.f;
      if (e < DEG_) {
        const int j = (e < n) ? e : e + 1;
        const int col = b * N_ + j;
        val = s_q[h] + QP[(size_t)col * 128 + 64 + h]
            + s_dr[e * 8 + 6] * s_w1t[0 * HID_ + h]
            + s_dr[e * 8 + 7] * s_w1t[1 * HID_ + h]
            + s_w1t[10 * HID_ + h];
#pragma unroll
        for (int f = 0; f < EF_; ++f)
          val += s_ea[e * EF_ + f] * s_w1t[(2 + f) * HID_ + h];
        val = silu_f(val);
      }
      s_a1[e * LDA_ + h] = val;          // pad rows 99..111 zeroed
    }
  }
  __syncthreads();

  // Fixed output-column tile per wave: B fragments preloaded to VGPRs once,
  // reused across all M-tiles (inner loop = 1 ds_load_b64 + 1 wmma).
  const int tn = wave & 3;
  const int nc = tn * 16 + l16;
  const int tm0 = wave >> 2;             // waves 0-3: tm 0,2,4,6; 4-7: 1,3,5

  // ---- GEMM1 (WMMA f32): m = silu(a1 @ W2 + b2) -> s_m ----
  {
    v2f bw[16];
#pragma unroll
    for (int ks = 0; ks < 16; ++ks) {
      const int ka = ks * 4 + half * 2;
      bw[ks].x = s_w2[ka * HID_ + nc];
      bw[ks].y = s_w2[(ka + 1) * HID_ + nc];
    }
    const float bias = b2[nc];
    for (int tm = tm0; tm < 7; tm += 2) {
      v8f acc = {0.f, 0.f, 0.f, 0.f, 0.f, 0.f, 0.f, 0.f};
      const float* aRow = s_a1 + (tm * 16 + l16) * LDA_;
#pragma unroll
      for (int ks = 0; ks < 16; ++ks) {
        const int ka = ks * 4 + half * 2;
        v2f a;
        a.x = aRow[ka]; a.y = aRow[ka + 1];
        acc = __builtin_amdgcn_wmma_f32_16x16x4_f32(false, a, false, bw[ks],
                                                    (short)0, acc, false, false);
      }
#pragma unroll
      for (int r = 0; r < 8; ++r)
        s_m[(tm * 16 + r + 8 * half) * LDA_ + nc] = silu_f(acc[r] + bias);
    }
  }
  __syncthreads();

  // ---- GEMM2 (WMMA f32): ch = silu(m @ C1 + cb1) -> s_a1 (reuse) ----
  {
    v2f bw[16];
#pragma unroll
    for (int ks = 0; ks < 16; ++ks) {
      const int ka = ks * 4 + half * 2;
      bw[ks].x = s_c1[ka * HID_ + nc];
      bw[ks].y = s_c1[(ka + 1) * HID_ + nc];
    }
    const float bias = cb1[nc];
    for (int tm = tm0; tm < 7; tm += 2) {
      v8f acc = {0.f, 0.f, 0.f, 0.f, 0.f, 0.f, 0.f, 0.f};
      const float* aRow = s_m + (tm * 16 + l16) * LDA_;
#pragma unroll
      for (int ks = 0; ks < 16; ++ks) {
        const int ka = ks * 4 + half * 2;
        v2f a;
        a.x = aRow[ka]; a.y = aRow[ka + 1];
        acc = __builtin_amdgcn_wmma_f32_16x16x4_f32(false, a, false, bw[ks],
                                                    (short)0, acc, false, false);
      }
#pragma unroll
      for (int r = 0; r < 8; ++r)
        s_a1[(tm * 16 + r + 8 * half) * LDA_ + nc] = silu_f(acc[r] + bias);
    }
  }
  __syncthreads();

  // ---- coord head c = ch @ C2 + cb2, equivariant trans accumulation ----
  if (tid < 2 * DEG_) {
    const int e = tid >> 1, k = tid & 1;
    float c = cb2[k];
    for (int r = 0; r < HID_; ++r) {     // rotated index: conflict-free LDS
      const int rr = (r + e) & 63;
      c += s_a1[e * LDA_ + rr] * C2[rr * 2 + k];
    }
#pragma unroll
    for (int d = 0; d < 3; ++d)          // coords_weight = 1.0
      atomicAdd(&s_xacc[d * 2 + k], s_dr[e * 8 + d * 2 + k] * c);
  }
  // agg = segment_sum(m) over this node's contiguous 99 edges
  if (tid < HID_) {
    float s = 0.f;
    for (int e = 0; e < DEG_; ++e) s += s_m[e * LDA_ + tid];
    s_agg[tid] = s;
  }
  __syncthreads();

  // ---- node MLP: hh' = silu([hh|agg] @ nW1 + nb1) @ nW2 + nb2 ----
  if (tid < HID_) {
    float v = nb1[tid];
    for (int r = 0; r < HID_; ++r) v += s_hh [r] * nW1[r * HID_ + tid];
    for (int r = 0; r < HID_; ++r) v += s_agg[r] * nW1[(HID_ + r) * HID_ + tid];
    s_u[tid] = silu_f(v);
  }
  __syncthreads();
  if (tid < HID_) {
    float o = nb2[tid];
    for (int r = 0; r < HID_; ++r) o += s_u[r] * nW2[r * HID_ + tid];
    hh_out[(size_t)i * HID_ + tid] = o;
  }
  if (tid >= HID_ && tid < HID_ + 6) {
    const int d = tid - HID_;
    x_out[(size_t)i * 6 + d] = s_xi[d] + s_xacc[d];
  }
}

// ---------------------------------------------------------------------------
extern "C" void kernel_launch(void* const* d_in, const int* in_sizes, int n_in,
                              void* d_out, int out_size, void* d_ws,
                              size_t ws_size, hipStream_t stream) {
  (void)in_sizes; (void)n_in; (void)out_size; (void)ws_size;
  const float* hfeat = (const float*)d_in[0];
  // d_in[1] = edges (int32) is ignored by the module (fully-connected graph)
  const float* rot   = (const float*)d_in[2];
  const float* eattr = (const float*)d_in[3];
  const float* embW  = (const float*)d_in[4];
  const float* embB  = (const float*)d_in[5];
  const float* eW1   = (const float*)d_in[6];
  const float* eb1   = (const float*)d_in[7];
  const float* eW2   = (const float*)d_in[8];
  const float* eb2   = (const float*)d_in[9];
  const float* nW1   = (const float*)d_in[10];
  const float* nb1   = (const float*)d_in[11];
  const float* nW2   = (const float*)d_in[12];
  const float* nb2   = (const float*)d_in[13];
  const float* cW1   = (const float*)d_in[14];
  const float* cb1   = (const float*)d_in[15];
  const float* cW2   = (const float*)d_in[16];
  const float* cb2   = (const float*)d_in[17];
  float* out = (float*)d_out;

  // Workspace: hhA|hhB (6400x64 each), xA|xB (6400x6 each), QP (6400x128)
  float* ws  = (float*)d_ws;
  float* hhA = ws; ws += (size_t)NN_ * HID_;
  float* hhB = ws; ws += (size_t)NN_ * HID_;
  float* xA  = ws; ws += (size_t)NN_ * 6;
  float* xB  = ws; ws += (size_t)NN_ * 6;
  float* QP  = ws;                       // NN_ * 128

  egnn_init<<<NN_, 64, 0, stream>>>(hfeat, rot, embW, embB, hhA, xA);

  for (int l = 0; l < L_; ++l) {
    const float* hh_in = (l & 1) ? hhB : hhA;
    const float* x_in  = (l & 1) ? xB  : xA;
    float* hh_out = (l == L_ - 1) ? out                      : ((l & 1) ? hhA : hhB);
    float* x_out  = (l == L_ - 1) ? out + (size_t)NN_ * HID_ : ((l & 1) ? xA  : xB);

    egnn_proj<<<NN_ / 16, 256, 0, stream>>>(hh_in, eW1 + (size_t)l * 138 * HID_, QP);
    egnn_layer<<<NN_, 256, 0, stream>>>(
        hh_in, x_in, QP, eattr,
        eW1 + (size_t)l * 138 * HID_, eb1 + (size_t)l * HID_,
        eW2 + (size_t)l * HID_ * HID_, eb2 + (size_t)l * HID_,
        nW1 + (size_t)l * 2 * HID_ * HID_, nb1 + (size_t)l * HID_,
        nW2 + (size_t)l * HID_ * HID_, nb2 + (size_t)l * HID_,
        cW1 + (size_t)l * HID_ * HID_, cb1 + (size_t)l * HID_,
        cW2 + (size_t)l * HID_ * 2, cb2 + (size_t)l * 2,
        hh_out, x_out);
  }
}